// GCNNode_73332271612103
// MI455X (gfx1250) — compile-verified
//
#include <hip/hip_runtime.h>
#include <math.h>

typedef float v2f __attribute__((ext_vector_type(2)));
typedef float v8f __attribute__((ext_vector_type(8)));

#define IN_DIM 128   // also HID_DIM

// ---------------- small elementwise kernels ----------------

__global__ void k_fill1(float* __restrict__ p, int n) {
    int i = blockIdx.x * blockDim.x + threadIdx.x;
    if (i < n) p[i] = 1.0f;                    // self-loop contributes 1 to degree
}

__global__ void k_degcount(const int* __restrict__ dst, float* __restrict__ deg, int E) {
    int i = blockIdx.x * blockDim.x + threadIdx.x;
    if (i < E) atomicAdd(&deg[dst[i]], 1.0f);
}

__global__ void k_rsqrt(float* __restrict__ deg, int n) {
    int i = blockIdx.x * blockDim.x + threadIdx.x;
    if (i < n) deg[i] = 1.0f / sqrtf(deg[i]);  // fp32 deg^-0.5
}

// buf[seg*DIM + d] = bias[d]  (pre-load bias so scatter-adds land on top of it)
template <int DIM>
__global__ void k_bias_init(float* __restrict__ buf, const float* __restrict__ bias,
                            int total) {
    int g = blockIdx.x * blockDim.x + threadIdx.x;
    if (g < total) buf[g] = bias[g % DIM];
}

__global__ void k_relu(float* __restrict__ p, int total) {
    int g = blockIdx.x * blockDim.x + threadIdx.x;
    if (g < total) { float v = p[g]; p[g] = v > 0.0f ? v : 0.0f; }
}

// ---------------- fp32 WMMA GEMM: C[n x OUTD] = A[n x 128] @ B[128 x OUTD] ----------------
// One wave32 computes one 16x16 output tile via 32x V_WMMA_F32_16X16X4_F32 (K=128).
// VGPR layouts per CDNA5 ISA 7.12.2:
//   A 16x4 : lane L holds A[M=L%16][K = 2*(L/16) + {0,1}] in the 2 source VGPRs
//   B 4x16 : lane L holds B[K = 2*(L/16) + {0,1}][N=L%16]
//   C 16x16: VGPR v of lane L holds C[M = v + 8*(L/16)][N=L%16]
// For OUTD=40 the 3rd col-tile over-reads: we CLAMP the B column instead of
// branching — lane L's B value only affects output column L%16, and columns
// >= OUTD are never stored, so clamped (garbage) values are harmless. This
// keeps the hot loop free of exec-mask diamonds.
template <int OUTD>
__global__ void k_gemm_wmma(const float* __restrict__ A, const float* __restrict__ B,
                            float* __restrict__ C, int n, int tilesN, int totalTiles) {
    int lane = threadIdx.x & 31;
    int wid  = (blockIdx.x * blockDim.x + threadIdx.x) >> 5;
    if (wid >= totalTiles) return;             // wave-uniform: EXEC stays all-ones for WMMA

    int tM = wid / tilesN;
    int tN = wid - tM * tilesN;

    int row   = tM * 16 + (lane & 15);
    int colB  = tN * 16 + (lane & 15);
    int khalf = (lane >> 4) << 1;              // 0 or 2
    bool colOK = (OUTD % 16 == 0) || (colB < OUTD);
    int  colLd = colOK ? colB : 0;             // clamped load column (see note above)

    const float* aptr = A + (size_t)row * IN_DIM + khalf;   // 8B-aligned (khalf even)
    const float* bptr = B + (size_t)khalf * OUTD + colLd;

    v8f c = {};
    #pragma unroll
    for (int kb = 0; kb < IN_DIM; kb += 4) {   // compile-time kb -> immediate offsets
        v2f a = *(const v2f*)(aptr + kb);
        v2f b;
        b.x = bptr[(size_t)kb * OUTD];
        b.y = bptr[(size_t)(kb + 1) * OUTD];
        c = __builtin_amdgcn_wmma_f32_16x16x4_f32(
                /*neg_a=*/false, a, /*neg_b=*/false, b,
                /*c_mod=*/(short)0, c, /*reuse_a=*/false, /*reuse_b=*/false);
    }

    int half = lane >> 4;
    #pragma unroll
    for (int v = 0; v < 8; ++v) {
        int m = tM * 16 + v + 8 * half;
        if (m < n && colOK) C[(size_t)m * OUTD + colB] = c[v];
    }
}

// ---------------- normalized scatter-add over edges + self-loops ----------------
// seg < E : agg[dst[seg]*DIM + d] += h[src[seg]*DIM + d] * dinv[src]*dinv[dst]
// seg >= E: self-loop for node (seg-E) with norm = dinv^2
template <int DIM>
__global__ void k_scatter(const int* __restrict__ src, const int* __restrict__ dst,
                          const float* __restrict__ dinv, const float* __restrict__ h,
                          float* __restrict__ agg, int E, int total) {
    int g = blockIdx.x * blockDim.x + threadIdx.x;
    if (g >= total) return;
    int seg = g / DIM;                          // DIM compile-time: shift or magic-mul
    int d   = g - seg * DIM;
    int s, t; float nrm;
    if (seg < E) {
        s = src[seg]; t = dst[seg];
        nrm = dinv[s] * dinv[t];
    } else {
        s = t = seg - E;
        float di = dinv[s];
        nrm = di * di;
    }
    atomicAdd(&agg[(size_t)t * DIM + d], h[(size_t)s * DIM + d] * nrm);
}

// ---------------- host-side launch ----------------

extern "C" void kernel_launch(void* const* d_in, const int* in_sizes, int n_in,
                              void* d_out, int out_size, void* d_ws, size_t ws_size,
                              hipStream_t stream) {
    const float* x  = (const float*)d_in[0];
    const int*   e  = (const int*)  d_in[1];
    const float* W1 = (const float*)d_in[2];
    const float* b1 = (const float*)d_in[3];
    const float* W2 = (const float*)d_in[4];
    const float* b2 = (const float*)d_in[5];
    float* out = (float*)d_out;

    const int N = in_sizes[0] / IN_DIM;
    const int E = in_sizes[1] / 2;
    constexpr int C = 40;                      // NUM_CLASSES

    const int* src = e;        // e[0, :]
    const int* dst = e + E;    // e[1, :]

    // workspace layout (fp32): dinv[N] | agg1[N*128] | hbuf[N*128] (H1, reused as H2 stride 40)
    float* ws   = (float*)d_ws;
    float* dinv = ws;
    float* agg1 = ws + (size_t)N;
    float* hbuf = agg1 + (size_t)N * IN_DIM;
    (void)ws_size; (void)n_in; (void)out_size;

    const int TB = 256;
    auto blk = [](int total, int tb) { return (total + tb - 1) / tb; };

    // 1) symmetric-norm coefficients: dinv = (1 + indeg)^-0.5
    k_fill1   <<<blk(N, TB), TB, 0, stream>>>(dinv, N);
    k_degcount<<<blk(E, TB), TB, 0, stream>>>(dst, dinv, E);
    k_rsqrt   <<<blk(N, TB), TB, 0, stream>>>(dinv, N);

    // 2) layer 1: H1 = X @ W1 (WMMA f32), agg1 = b1 + scatter(norm * H1[src] -> dst), ReLU
    {
        int tilesM = (N + 15) / 16, tilesN = IN_DIM / 16;     // 6250 x 8
        int totalTiles = tilesM * tilesN;
        k_gemm_wmma<IN_DIM><<<blk(totalTiles * 32, TB), TB, 0, stream>>>(
            x, W1, hbuf, N, tilesN, totalTiles);
    }
    {
        int total = N * IN_DIM;
        k_bias_init<IN_DIM><<<blk(total, TB), TB, 0, stream>>>(agg1, b1, total);
        int stotal = (E + N) * IN_DIM;
        k_scatter<IN_DIM><<<blk(stotal, TB), TB, 0, stream>>>(src, dst, dinv, hbuf, agg1, E, stotal);
        k_relu    <<<blk(total, TB), TB, 0, stream>>>(agg1, total);
    }

    // 3) layer 2: H2 = H @ W2 (WMMA f32, 3 col-tiles, clamped loads to 40), out = b2 + scatter
    {
        int tilesM = (N + 15) / 16, tilesN = (C + 15) / 16;   // 6250 x 3
        int totalTiles = tilesM * tilesN;
        k_gemm_wmma<C><<<blk(totalTiles * 32, TB), TB, 0, stream>>>(
            agg1, W2, hbuf, N, tilesN, totalTiles);
    }
    {
        int total = N * C;
        k_bias_init<C><<<blk(total, TB), TB, 0, stream>>>(out, b2, total);
        int stotal = (E + N) * C;
        k_scatter<C><<<blk(stotal, TB), TB, 0, stream>>>(src, dst, dinv, hbuf, out, E, stotal);
    }
}